// Spe_loss_7258494730591
// MI455X (gfx1250) — compile-verified
//
#include <hip/hip_runtime.h>

// Problem constants (B=16, C=64, H=W=144)
#define BC      1024          // reduction length per spatial position
#define HW      20736         // 144*144 spatial positions
#define HW4     5184          // HW / 4 (float4 positions)
#define NCHUNK  16            // bc-chunks (partial-sum split for parallelism)
#define CHUNK   64            // bc planes per chunk  (NCHUNK*CHUNK == BC)
#define NTILE   1296          // HW / 16 position tiles for WMMA finalize
#define FBLOCKS 162           // NTILE / 8 waves-per-block

typedef __attribute__((ext_vector_type(2))) float v2f;
typedef __attribute__((ext_vector_type(8))) float v8f;

// ---------------------------------------------------------------------------
// Pass 1: HBM-bound streaming reduction. Each thread owns 4 consecutive
// spatial positions (float4 = one global_load_b128) and one bc-chunk of 64
// planes. Writes f32 partial sums (fz, xx, yy) to workspace, each element
// written exactly once -> deterministic, no init needed.
// ---------------------------------------------------------------------------
__global__ __launch_bounds__(256) void spe_partial_kernel(
    const float* __restrict__ x, const float* __restrict__ y,
    float* __restrict__ pz, float* __restrict__ px, float* __restrict__ py) {
  const int p = blockIdx.x * blockDim.x + threadIdx.x;  // float4 position
  if (p >= HW4) return;
  const int c = blockIdx.y;                             // bc-chunk
  const float4* __restrict__ x4 = (const float4*)x;
  const float4* __restrict__ y4 = (const float4*)y;

  float4 fz = make_float4(0.f, 0.f, 0.f, 0.f);
  float4 xx = make_float4(0.f, 0.f, 0.f, 0.f);
  float4 yy = make_float4(0.f, 0.f, 0.f, 0.f);

  size_t base = (size_t)c * CHUNK * HW4 + (size_t)p;
#pragma unroll 4
  for (int k = 0; k < CHUNK; ++k) {
    float4 xv = x4[base + (size_t)k * HW4];
    float4 yv = y4[base + (size_t)k * HW4];
    fz.x += xv.x * yv.x;  fz.y += xv.y * yv.y;
    fz.z += xv.z * yv.z;  fz.w += xv.w * yv.w;
    xx.x += xv.x * xv.x;  xx.y += xv.y * xv.y;
    xx.z += xv.z * xv.z;  xx.w += xv.w * xv.w;
    yy.x += yv.x * yv.x;  yy.y += yv.y * yv.y;
    yy.z += yv.z * yv.z;  yy.w += yv.w * yv.w;
  }
  const size_t o = (size_t)c * HW4 + (size_t)p;
  ((float4*)pz)[o] = fz;
  ((float4*)px)[o] = xx;
  ((float4*)py)[o] = yy;
}

// ---------------------------------------------------------------------------
// Pass 2: combine the 16 chunk-partials per position with the matrix pipe.
// A = all-ones 16x4 f32, B = 4 chunk rows x 16 positions -> every row of
// D = A*B + C is the column sum, so 4 accumulating V_WMMA_F32_16X16X4_F32
// steps reduce all 16 chunks. All-ones A makes the result independent of the
// exact B lane striping (any column-preserving layout sums the same values).
// Then acos + wave/LDS reduction -> one partial per block.
// ---------------------------------------------------------------------------
__global__ __launch_bounds__(256) void spe_finalize_kernel(
    const float* __restrict__ pz, const float* __restrict__ px,
    const float* __restrict__ py, float* __restrict__ blockout) {
  const int tid  = threadIdx.x;
  const int lane = tid & 31;
  const int wave = tid >> 5;
  const int tile = blockIdx.x * 8 + wave;       // 16 positions per wave
  const int col  = tile * 16 + (lane & 15);     // N index (position)
  const int k0   = (lane < 16) ? 0 : 2;         // K rows this lane supplies

  const v2f ones = {1.0f, 1.0f};
  v8f az = {}; v8f ax = {}; v8f ay = {};

#pragma unroll
  for (int kb = 0; kb < NCHUNK; kb += 4) {
    v2f bz, bx, by;
    size_t r0 = (size_t)(kb + k0)     * HW + (size_t)col;
    size_t r1 = (size_t)(kb + k0 + 1) * HW + (size_t)col;
    bz.x = pz[r0]; bz.y = pz[r1];
    bx.x = px[r0]; bx.y = px[r1];
    by.x = py[r0]; by.y = py[r1];
    az = __builtin_amdgcn_wmma_f32_16x16x4_f32(false, ones, false, bz,
                                               (short)0, az, false, false);
    ax = __builtin_amdgcn_wmma_f32_16x16x4_f32(false, ones, false, bx,
                                               (short)0, ax, false, false);
    ay = __builtin_amdgcn_wmma_f32_16x16x4_f32(false, ones, false, by,
                                               (short)0, ay, false, false);
  }

  // Row M=0 of D lives in acc[0]; all rows are identical (A is all-ones),
  // so lanes 16..31 duplicate lanes 0..15 — mask them out of the sum.
  float ang = 0.0f;
  if (lane < 16) {
    float fz = az[0], xx = ax[0], yy = ay[0];
    float r = fz / (sqrtf(xx) * sqrtf(yy));
    r = fminf(fmaxf(r, -1.0f), 1.0f);
    ang = acosf(r);
  }
#pragma unroll
  for (int off = 16; off > 0; off >>= 1) ang += __shfl_down(ang, off);

  __shared__ float wsum[8];
  if (lane == 0) wsum[wave] = ang;
  __syncthreads();
  if (tid == 0) {
    float s = 0.0f;
#pragma unroll
    for (int i = 0; i < 8; ++i) s += wsum[i];
    blockout[blockIdx.x] = s;
  }
}

// ---------------------------------------------------------------------------
// Pass 3: single-block sum of block partials, scale by 1/144^2.
// ---------------------------------------------------------------------------
__global__ __launch_bounds__(256) void spe_sum_kernel(
    const float* __restrict__ blockout, int n, float* __restrict__ out) {
  __shared__ float s[256];
  float v = (threadIdx.x < n) ? blockout[threadIdx.x] : 0.0f;
  s[threadIdx.x] = v;
  __syncthreads();
#pragma unroll
  for (int off = 128; off > 0; off >>= 1) {
    if (threadIdx.x < off) s[threadIdx.x] += s[threadIdx.x + off];
    __syncthreads();
  }
  if (threadIdx.x == 0) out[0] = s[0] * (1.0f / (float)HW);
}

extern "C" void kernel_launch(void* const* d_in, const int* in_sizes, int n_in,
                              void* d_out, int out_size, void* d_ws, size_t ws_size,
                              hipStream_t stream) {
  const float* x = (const float*)d_in[0];
  const float* y = (const float*)d_in[1];
  // d_in[2] is the Python scalar 'shape' (=144); baked into the constants.

  float* pz = (float*)d_ws;                 // [NCHUNK][HW]
  float* px = pz + (size_t)NCHUNK * HW;     // [NCHUNK][HW]
  float* py = px + (size_t)NCHUNK * HW;     // [NCHUNK][HW]
  float* bo = py + (size_t)NCHUNK * HW;     // [FBLOCKS]

  dim3 grid1((HW4 + 255) / 256, NCHUNK);
  spe_partial_kernel<<<grid1, 256, 0, stream>>>(x, y, pz, px, py);
  spe_finalize_kernel<<<FBLOCKS, 256, 0, stream>>>(pz, px, py, bo);
  spe_sum_kernel<<<1, 256, 0, stream>>>(bo, FBLOCKS, (float*)d_out);
}